// CrfDecodeForwardRnnCell_47175920779504
// MI455X (gfx1250) — compile-verified
//
#include <hip/hip_runtime.h>
#include <hip/hip_bf16.h>

typedef __attribute__((ext_vector_type(2))) float v2f;
typedef __attribute__((ext_vector_type(8))) float v8f;

#define BATCH 1024
#define SEQL  512
#define NTAGS 48

// One wave32 per batch row. 48x48 = 3x3 tiles of 16x16.
// WMMA computes state[i] + trans[i][j] as rank-1 update: D = A(state)*B(e0) + C(trans).
// VALU tree-argmax over rows; one ds_swizzle half-merge per j-tile with exact
// index tie-break, so the merged result is valid in all 32 lanes.
__global__ __launch_bounds__(256) void crf_viterbi_wmma(
    const float* __restrict__ pot,    // [B, T, 48]
    const float* __restrict__ trans,  // [48, 48]
    int*  __restrict__ bp,            // [B, T-1, 48]
    float* __restrict__ scores)       // [B, 48]
{
  const int lane = threadIdx.x & 31;
  const int wave = threadIdx.x >> 5;
  const int b    = blockIdx.x * 8 + wave;
  const int half = lane >> 4;     // 0: rows 0-7 of tile, 1: rows 8-15
  const int l16  = lane & 15;     // column within tile
  const int rowBase = half * 8;

  // Transition matrix tiles held in registers in WMMA C/D layout:
  // VGPR r, lanes 0-15: M=r, N=l16 ; lanes 16-31: M=8+r, N=l16
  v8f c[3][3];
#pragma unroll
  for (int it = 0; it < 3; ++it)
#pragma unroll
    for (int jt = 0; jt < 3; ++jt)
#pragma unroll
      for (int r = 0; r < 8; ++r)
        c[it][jt][r] = trans[(it * 16 + rowBase + r) * NTAGS + jt * 16 + l16];

  // B operand = e0 row of ones (4x16): VGPR0 lanes0-15 (K=0 row) -> 1.0,
  // lanes16-31 (K=1 row) -> 0.0 ; VGPR1 (K=2,3 rows) -> 0.0
  v2f bop;
  bop.x = (half == 0) ? 1.0f : 0.0f;
  bop.y = 0.0f;

  const float* potb = pot + (size_t)b * SEQL * NTAGS;

  // Initial state = potentials[:,0]; replicated into both lane halves.
  float s0 = potb[l16];
  float s1 = potb[16 + l16];
  float s2 = potb[32 + l16];

  int* bpb = bp + (size_t)b * (SEQL - 1) * NTAGS;

#pragma unroll 2
  for (int t = 1; t < SEQL; ++t) {
    const float* pt = potb + (size_t)t * NTAGS;
    if (t + 8 < SEQL) __builtin_prefetch(pt + 8 * NTAGS, 0, 1);  // global_prefetch

    float sIn[3] = { s0, s1, s2 };
    float ns[3];
    int   bi[3];
#pragma unroll
    for (int jt = 0; jt < 3; ++jt) {
      float m = 0.0f;
      int   idx = 0;
#pragma unroll
      for (int it = 0; it < 3; ++it) {
        // A operand (16x4 f32): K=0 = state column. Hi lanes land in the K=2
        // slot, which multiplies B's zero row -> contributes exactly 0 (state
        // is always finite, so no Inf*0 hazard). No masking needed.
        v2f a;
        a.x = sIn[it];
        a.y = 0.0f;
        // D[m][n] = state[m]*1 + trans[it*16+m][jt*16+n]  (bit-exact fp32 add)
        v8f d = __builtin_amdgcn_wmma_f32_16x16x4_f32(
            false, a, false, bop, (short)0, c[it][jt], false, false);

        // Tree argmax over this half's 8 rows (strict '>' keeps first occurrence).
        bool g;
        g = d[1] > d[0]; float m01 = g ? d[1] : d[0]; int i01 = g ? 1 : 0;
        g = d[3] > d[2]; float m23 = g ? d[3] : d[2]; int i23 = g ? 3 : 2;
        g = d[5] > d[4]; float m45 = g ? d[5] : d[4]; int i45 = g ? 5 : 4;
        g = d[7] > d[6]; float m67 = g ? d[7] : d[6]; int i67 = g ? 7 : 6;
        g = m23 > m01;  float m03 = g ? m23 : m01;   int i03 = g ? i23 : i01;
        g = m67 > m45;  float m47 = g ? m67 : m45;   int i47 = g ? i67 : i45;
        g = m47 > m03;  float lm  = g ? m47 : m03;   int li  = g ? i47 : i03;
        li += rowBase + it * 16;  // global source-tag index

        // Accumulate i-tiles per lane-half (ascending contiguous row ranges,
        // strict '>' preserves first occurrence within the half).
        if (it == 0) { m = lm; idx = li; }
        else { bool g3 = lm > m; m = g3 ? lm : m; idx = g3 ? li : idx; }
      }
      // Single half-merge per j-tile via XOR-16 swizzle. Rows are interleaved
      // between halves, so break ties exactly by smaller source index.
      float om = __int_as_float(__builtin_amdgcn_ds_swizzle(__float_as_int(m), 0x401F));
      int   oi = __builtin_amdgcn_ds_swizzle(idx, 0x401F);
      bool better = (om > m) || ((om == m) && (oi < idx));
      m   = better ? om : m;
      idx = better ? oi : idx;
      // (m, idx) now valid in ALL 32 lanes.

      ns[jt] = pt[jt * 16 + l16] + m;  // new_state = unary + max
      bi[jt] = idx;
    }
    s0 = ns[0]; s1 = ns[1]; s2 = ns[2];

    // Backpointers: full-wave 128B store covers j=0..31, 16-lane store for j=32..47.
    int* o = bpb + (size_t)(t - 1) * NTAGS;
    o[lane] = half ? bi[1] : bi[0];
    if (half == 0) o[32 + l16] = bi[2];
  }

  float* so = scores + (size_t)b * NTAGS;
  so[lane] = half ? s1 : s0;
  if (half == 0) so[32 + l16] = s2;
}

extern "C" void kernel_launch(void* const* d_in, const int* in_sizes, int n_in,
                              void* d_out, int out_size, void* d_ws, size_t ws_size,
                              hipStream_t stream) {
  const float* pot   = (const float*)d_in[0];   // [1024, 512, 48] f32
  const float* trans = (const float*)d_in[1];   // [48, 48] f32
  int*   bp     = (int*)d_out;                                         // [1024, 511, 48] i32
  float* scores = (float*)d_out + (size_t)BATCH * (SEQL - 1) * NTAGS;  // [1024, 48] f32

  dim3 grid(BATCH / 8);   // 8 waves (8 batch rows) per 256-thread block
  dim3 block(256);
  hipLaunchKernelGGL(crf_viterbi_wmma, grid, block, 0, stream, pot, trans, bp, scores);
}